// GraphSAGE_72997264162856
// MI455X (gfx1250) — compile-verified
//
#include <hip/hip_runtime.h>
#include <hip/hip_bf16.h>

typedef __attribute__((ext_vector_type(2))) float v2f;
typedef __attribute__((ext_vector_type(8))) float v8f;

// ---------------------------------------------------------------------------
// Zero a float region (grid-stride).
// ---------------------------------------------------------------------------
__global__ void sage_zero_kernel(float* __restrict__ p, size_t count) {
    size_t i = blockIdx.x * (size_t)blockDim.x + threadIdx.x;
    size_t stride = (size_t)gridDim.x * blockDim.x;
    for (; i < count; i += stride) p[i] = 0.0f;
}

// ---------------------------------------------------------------------------
// deg[dst] += 1 for every edge.
// ---------------------------------------------------------------------------
__global__ void sage_degree_kernel(const long long* __restrict__ dst,
                                   float* __restrict__ deg, int e) {
    int i = blockIdx.x * blockDim.x + threadIdx.x;
    if (i < e) {
        long long d = dst[i];
        atomicAdd(deg + d, 1.0f);
    }
}

// ---------------------------------------------------------------------------
// sum[dst,:] += feat[src,:]   (64 features per edge; 16 threads/edge, float4
// gather + 4x global_atomic_add_f32 to consecutive addresses so the L2 atomic
// units can combine them).
// ---------------------------------------------------------------------------
__global__ void sage_scatter_kernel(const float* __restrict__ feat,
                                    const long long* __restrict__ src,
                                    const long long* __restrict__ dst,
                                    float* __restrict__ sum, int e) {
    int idx = blockIdx.x * blockDim.x + threadIdx.x;
    int total = e << 4;               // e * 16 float4 chunks
    if (idx >= total) return;
    int ed = idx >> 4;
    int c  = (idx & 15) << 2;         // feature offset 0,4,...,60
    long long s = src[ed];
    long long d = dst[ed];
    const float4 v = *(const float4*)(feat + (size_t)s * 64 + c);
    float* p = sum + (size_t)d * 64 + c;
    atomicAdd(p + 0, v.x);
    atomicAdd(p + 1, v.y);
    atomicAdd(p + 2, v.z);
    atomicAdd(p + 3, v.w);
}

// ---------------------------------------------------------------------------
// Fused SAGE layer:  out = act( (sum/max(deg,1)) @ Wl + b + feat @ Wr )
// using V_WMMA_F32_16X16X4_F32 (full fp32 precision).
//
// Per-wave 16x16 output tile. VGPR layouts per CDNA5 ISA 7.12.2:
//   A (16x4 f32):  lane L holds A[L%16][k0+2*(L/16) + {0,1}] in 2 VGPRs
//   B (4x16 f32):  lane L holds B[k0+2*(L/16) + {0,1}][L%16]
//   C/D (16x16):   vgpr r, lane L -> D[r + 8*(L/16)][L%16]
// ---------------------------------------------------------------------------
template <int DIN, int DOUT, bool RELU>
__global__ void sage_wmma_kernel(const float* __restrict__ feat,
                                 const float* __restrict__ sum,
                                 const float* __restrict__ deg,
                                 const float* __restrict__ Wl,
                                 const float* __restrict__ Wr,
                                 const float* __restrict__ bias,
                                 float* __restrict__ out, int n) {
    constexpr int COLT = DOUT / 16;   // 16-wide col tiles per row group
    constexpr int RG   = 8 / COLT;    // row groups per 8-wave block
    constexpr int ROWS = RG * 16;     // rows handled per block

    const int wave  = threadIdx.x >> 5;
    const int lane  = threadIdx.x & 31;
    const int rg    = wave / COLT;
    const int n0    = (wave % COLT) * 16;
    const int m0    = blockIdx.x * ROWS + rg * 16;
    const int lrow  = lane & 15;
    const int khalf = lane >> 4;      // 0: K pair {0,1}; 1: K pair {2,3}

    int row = m0 + lrow;
    if (row >= n) row = n - 1;        // clamp (EXEC must stay all-ones for WMMA)
    const int col = n0 + lrow;

    const float invd = 1.0f / fmaxf(deg[row], 1.0f);
    const float* __restrict__ srow = sum  + (size_t)row * DIN;
    const float* __restrict__ frow = feat + (size_t)row * DIN;

    v8f c = {};

    // mean @ Wl
#pragma unroll
    for (int k0 = 0; k0 < DIN; k0 += 4) {
        const int kk = k0 + 2 * khalf;
        v2f a, b;
        a.x = srow[kk] * invd;
        a.y = srow[kk + 1] * invd;
        b.x = Wl[(size_t)kk * DOUT + col];
        b.y = Wl[(size_t)(kk + 1) * DOUT + col];
        c = __builtin_amdgcn_wmma_f32_16x16x4_f32(false, a, false, b,
                                                  (short)0, c, false, false);
    }
    // feat @ Wr
#pragma unroll
    for (int k0 = 0; k0 < DIN; k0 += 4) {
        const int kk = k0 + 2 * khalf;
        v2f a, b;
        a.x = frow[kk];
        a.y = frow[kk + 1];
        b.x = Wr[(size_t)kk * DOUT + col];
        b.y = Wr[(size_t)(kk + 1) * DOUT + col];
        c = __builtin_amdgcn_wmma_f32_16x16x4_f32(false, a, false, b,
                                                  (short)0, c, false, false);
    }

    const float bv  = bias[col];
    const int   mrb = 8 * khalf;
#pragma unroll
    for (int r = 0; r < 8; ++r) {
        int m = m0 + mrb + r;
        if (m < n) {
            float v = c[r] + bv;
            if (RELU) v = fmaxf(v, 0.0f);
            out[(size_t)m * DOUT + col] = v;
        }
    }
}

// ---------------------------------------------------------------------------
// Launch
// ---------------------------------------------------------------------------
extern "C" void kernel_launch(void* const* d_in, const int* in_sizes, int n_in,
                              void* d_out, int out_size, void* d_ws, size_t ws_size,
                              hipStream_t stream) {
    const float*     x    = (const float*)d_in[0];
    const long long* ei   = (const long long*)d_in[1];   // int64 (2, E)
    const float*     W1_l = (const float*)d_in[2];
    const float*     W1_r = (const float*)d_in[3];
    const float*     b1   = (const float*)d_in[4];
    const float*     W2_l = (const float*)d_in[5];
    const float*     W2_r = (const float*)d_in[6];
    const float*     b2   = (const float*)d_in[7];
    float*           out  = (float*)d_out;

    const int n = in_sizes[0] / 64;     // 100000
    const int e = in_sizes[1] / 2;      // 1200000
    const long long* src = ei;
    const long long* dst = ei + e;

    // workspace layout: deg [nAlign] | sum [n*64] | h [n*64]
    const int nAlign = (n + 255) & ~255;
    float* deg = (float*)d_ws;
    float* sum = deg + nAlign;
    float* h   = sum + (size_t)n * 64;

    const int ZB = 1024, TB = 256;
    const int scatterBlocks = ((e << 4) + TB - 1) / TB;

    // ---- layer 1 ----
    sage_zero_kernel<<<ZB, TB, 0, stream>>>(deg, (size_t)nAlign + (size_t)n * 64);
    sage_degree_kernel<<<(e + TB - 1) / TB, TB, 0, stream>>>(dst, deg, e);
    sage_scatter_kernel<<<scatterBlocks, TB, 0, stream>>>(x, src, dst, sum, e);
    sage_wmma_kernel<64, 64, true><<<(n + 31) / 32, TB, 0, stream>>>(
        x, sum, deg, W1_l, W1_r, b1, h, n);

    // ---- layer 2 ----
    sage_zero_kernel<<<ZB, TB, 0, stream>>>(sum, (size_t)n * 64);
    sage_scatter_kernel<<<scatterBlocks, TB, 0, stream>>>(h, src, dst, sum, e);
    sage_wmma_kernel<64, 32, false><<<(n + 63) / 64, TB, 0, stream>>>(
        h, sum, deg, W2_l, W2_r, b2, out, n);
}